// WNOConv2d_fast_17978733101702
// MI455X (gfx1250) — compile-verified
//
#include <hip/hip_runtime.h>
#include <hip/hip_bf16.h>

// ---------------------------------------------------------------------------
// WNO conv2d: db4 DWT -> per-mode 64x64 channel mixing (fp32 WMMA) -> IDWT
// MI455X reasoning: weight stream (294 MB) dominates -> memory bound; keep
// fp32 end-to-end and use V_WMMA_F32_16X16X4_F32 for the per-mode GEMMs
// (M=16 batch, N=64 Cout, K=64 Cin  ->  4 waves x 16 chained K=4 WMMAs).
// Round 2: stage the shared A tile (16x64 coeff block) in LDS once per block
// to kill the 4x redundant global reads across the 4 waves.
// ---------------------------------------------------------------------------

#define Bsz   16
#define Csz   64
#define Hsz   128
#define Wsz   128
#define Msz   67            // wavelet modes per axis
#define M2    (Msz*Msz)     // 4489
#define REGION_FLOATS (4u*M2*Csz*Bsz)   // 18,386,944 floats = 73.55 MB

__constant__ float DEC_LO[8] = {
  -0.010597401784997278f,  0.032883011666982945f,  0.030841381835986965f,
  -0.18703481171888114f,  -0.02798376941698385f,   0.6308807679295904f,
   0.7148465705525415f,    0.23037781330885523f };
__constant__ float DEC_HI[8] = {
  -0.23037781330885523f,   0.7148465705525415f,   -0.6308807679295904f,
  -0.02798376941698385f,   0.18703481171888114f,   0.030841381835986965f,
  -0.032883011666982945f, -0.010597401784997278f };

typedef __attribute__((ext_vector_type(2))) float v2f;
typedef __attribute__((ext_vector_type(8))) float v8f;

__device__ __forceinline__ int reflect128(int q) {
  // jnp.pad mode='symmetric' (edge-including reflection); single bounce is
  // enough for pad<=6 on N=128.
  q = (q < 0) ? (-1 - q) : q;
  q = (q >= Hsz) ? (2*Hsz - 1 - q) : q;
  return q;
}

// ---------------------------------------------------------------------------
// Stage 1: analysis along W.  x (B,C,H,W) -> lohi [b][c][bw][y][j]
// out[j] = sum_t DEC[7-t] * x_pad[2j + t - 6]   (correlation w/ reversed filt)
// ---------------------------------------------------------------------------
__global__ void dwt_rows(const float* __restrict__ x, float* __restrict__ lohi) {
  int tid = blockIdx.x * blockDim.x + threadIdx.x;
  const int total = Bsz*Csz*Hsz*Msz;                    // 8,781,824
  if (tid >= total) return;
  int j = tid % Msz;
  int y = (tid / Msz) % Hsz;
  int c = (tid / (Msz*Hsz)) % Csz;
  int b =  tid / (Msz*Hsz*Csz);
  const float* row = x + (((size_t)b*Csz + c)*Hsz + y)*Wsz;
  float lo = 0.f, hi = 0.f;
  #pragma unroll
  for (int t = 0; t < 8; ++t) {
    float v = row[reflect128(2*j + t - 6)];
    lo += DEC_LO[7-t] * v;
    hi += DEC_HI[7-t] * v;
  }
  size_t o = ((((size_t)b*Csz + c)*2 + 0)*Hsz + y)*Msz + j;
  lohi[o] = lo;
  lohi[o + (size_t)Hsz*Msz] = hi;                       // bw=1 plane
}

// ---------------------------------------------------------------------------
// Stage 2: analysis along H.  lohi -> coeff[band][mode][i(C)][m(B)]
// band = bw*2 + bh   (0=ll/w1, 1=lh/w2, 2=hl/w3, 3=hh/w4)
// Layout chosen so WMMA A-fragments read contiguous 16-float runs.
// ---------------------------------------------------------------------------
__global__ void dwt_cols(const float* __restrict__ lohi, float* __restrict__ coeff) {
  int tid = blockIdx.x * blockDim.x + threadIdx.x;
  const int total = 2*M2*Csz*Bsz;                       // 9,193,472
  if (tid >= total) return;
  int b  =  tid        & 15;
  int c  = (tid >> 4)  & 63;
  int j  = (tid >> 10) % Msz;
  int yp = (tid / (1024*Msz)) % Msz;
  int bw =  tid / (1024*M2);
  const float* col = lohi + ((((size_t)b*Csz + c)*2 + bw)*Hsz)*Msz + j;
  float lo = 0.f, hi = 0.f;
  #pragma unroll
  for (int t = 0; t < 8; ++t) {
    float v = col[(size_t)reflect128(2*yp + t - 6) * Msz];
    lo += DEC_LO[7-t] * v;
    hi += DEC_HI[7-t] * v;
  }
  size_t mode = (size_t)yp*Msz + j;
  coeff[(((size_t)(bw*2+0)*M2 + mode)*Csz + c)*Bsz + b] = lo;
  coeff[(((size_t)(bw*2+1)*M2 + mode)*Csz + c)*Bsz + b] = hi;
}

// ---------------------------------------------------------------------------
// Per-mode channel mixing with fp32 WMMA.
// One block (128 thr = 4 waves) per (band, mode).  The 16x64 A tile (coeff)
// is staged once into LDS (coalesced b128 loads), then each wave computes
// its 16(batch) x 16(out-ch) tile [n0 = 16*wave] as 16 chained
// V_WMMA_F32_16X16X4_F32, reading A-fragments from LDS (ds_load) and the
// gathered fp32 weight fragments straight from the HBM stream.
// Fragment layouts per ISA 7.12.2 (fp32, wave32):
//   A 16x4 : lane l holds A[l%16][k0 + 2*(l/16) + v],  v = VGPR 0/1
//   B 4x16 : lane l holds B[k0 + 2*(l/16) + v][l%16]
//   C/D    : acc[r] = D[r + 8*(l/16)][l%16]
// Output: mixed[band][mode][m(B)][o(C)]  (coalesced along o).
// ---------------------------------------------------------------------------
__global__ void mix_wmma(const float* __restrict__ coeff,
                         const float* __restrict__ w1, const float* __restrict__ w2,
                         const float* __restrict__ w3, const float* __restrict__ w4,
                         float* __restrict__ mixed) {
  __shared__ float As[Csz*Bsz];                 // 1024 floats = 4 KB

  int band = blockIdx.x / M2;
  int mode = blockIdx.x % M2;
  const float* W = (band == 0) ? w1 : (band == 1) ? w2 : (band == 2) ? w3 : w4;

  const float* A = coeff + ((size_t)band*M2 + mode)*(Csz*Bsz);  // [i][b]

  // Cooperative A staging: 256 float4 = 128 threads x 2, fully coalesced.
  {
    const float4* Ain = (const float4*)A;
    float4* As4 = (float4*)As;
    As4[threadIdx.x]       = Ain[threadIdx.x];
    As4[threadIdx.x + 128] = Ain[threadIdx.x + 128];
  }
  __syncthreads();

  const int lane  = threadIdx.x & 31;
  const int wave  = threadIdx.x >> 5;     // 0..3 -> N-tile
  const int n0    = wave << 4;
  const int mrow  = lane & 15;            // A row / B col / D col
  const int khalf = (lane >> 4) << 1;     // 0 or 2
  const int ocol  = n0 + mrow;

  v8f acc = {};
  #pragma unroll
  for (int k0 = 0; k0 < 64; k0 += 4) {
    v2f a, bf;
    int k = k0 + khalf;
    a[0]  = As[(k    )*Bsz + mrow];
    a[1]  = As[(k + 1)*Bsz + mrow];
    bf[0] = W[((size_t)(k    )*Csz + ocol)*M2 + mode];
    bf[1] = W[((size_t)(k + 1)*Csz + ocol)*M2 + mode];
    acc = __builtin_amdgcn_wmma_f32_16x16x4_f32(
        /*neg_a=*/false, a, /*neg_b=*/false, bf,
        /*c_mod=*/(short)0, acc, /*reuse_a=*/false, /*reuse_b=*/false);
  }
  float* O = mixed + ((size_t)band*M2 + mode)*(Bsz*Csz);
  const int mhi = (lane >> 4) << 3;
  #pragma unroll
  for (int r = 0; r < 8; ++r)
    O[(size_t)(r + mhi)*Csz + n0 + mrow] = acc[r];
}

// ---------------------------------------------------------------------------
// IDWT stage 1: synthesis along H.
// mixed[band][mode][b][o] -> tmp[side][b][c][n(128)][x'(67)]
// side 0 = sfb(ll,lh), side 1 = sfb(hl,hh);  out[n] = sum over even-parity taps
// ---------------------------------------------------------------------------
__global__ void idwt_cols(const float* __restrict__ mixed, float* __restrict__ tmp) {
  int tid = blockIdx.x * blockDim.x + threadIdx.x;
  const int total = 2*Bsz*Csz*Hsz*Msz;                  // 17,563,648
  if (tid >= total) return;
  int xq =  tid % Msz;
  int n  = (tid / Msz) % Hsz;
  int c  = (tid / (Msz*Hsz)) % Csz;
  int b  = (tid / (Msz*Hsz*Csz)) % Bsz;
  int sd =  tid / (Msz*Hsz*Csz*Bsz);
  int b0 = sd * 2;
  float acc = 0.f;
  #pragma unroll
  for (int t = 0; t < 8; ++t) {
    int num = n + t - 1;
    if (num & 1) continue;                 // upsampled zeros (also kills num<0)
    int j = num >> 1;
    if (j >= Msz) continue;
    size_t mo = (size_t)j*Msz + xq;
    float v0 = mixed[(((size_t)(b0    )*M2 + mo)*Bsz + b)*Csz + c];
    float v1 = mixed[(((size_t)(b0 + 1)*M2 + mo)*Bsz + b)*Csz + c];
    acc += DEC_LO[t]*v0 + DEC_HI[t]*v1;
  }
  tmp[tid] = acc;   // layout == [side][b][c][n][x']
}

// ---------------------------------------------------------------------------
// IDWT stage 2: synthesis along W.  tmp -> out (B,C,H,W), coalesced stores.
// ---------------------------------------------------------------------------
__global__ void idwt_rows(const float* __restrict__ tmp, float* __restrict__ out) {
  int tid = blockIdx.x * blockDim.x + threadIdx.x;
  const int total = Bsz*Csz*Hsz*Wsz;                    // 16,777,216
  if (tid >= total) return;
  int w =  tid % Wsz;
  int n = (tid / Wsz) % Hsz;
  int c = (tid / (Wsz*Hsz)) % Csz;
  int b =  tid / (Wsz*Hsz*Csz);
  const float* lo = tmp + ((((size_t)0*Bsz + b)*Csz + c)*Hsz + n)*Msz;
  const float* hi = tmp + ((((size_t)1*Bsz + b)*Csz + c)*Hsz + n)*Msz;
  float acc = 0.f;
  #pragma unroll
  for (int t = 0; t < 8; ++t) {
    int num = w + t - 1;
    if (num & 1) continue;
    int j = num >> 1;
    if (j >= Msz) continue;
    acc += DEC_LO[t]*lo[j] + DEC_HI[t]*hi[j];
  }
  out[tid] = acc;
}

// ---------------------------------------------------------------------------
extern "C" void kernel_launch(void* const* d_in, const int* in_sizes, int n_in,
                              void* d_out, int out_size, void* d_ws, size_t ws_size,
                              hipStream_t stream) {
  const float* x  = (const float*)d_in[0];
  const float* w1 = (const float*)d_in[1];
  const float* w2 = (const float*)d_in[2];
  const float* w3 = (const float*)d_in[3];
  const float* w4 = (const float*)d_in[4];
  float* out = (float*)d_out;

  // Ping-pong regions (73.56 MB each):
  //   A: lohi (dwt tmp) -> mixed (post-GEMM)
  //   B: coeff (pre-GEMM) -> tmp (idwt intermediate)
  float* regionA = (float*)d_ws;
  float* regionB = regionA + (size_t)REGION_FLOATS;

  dwt_rows <<<(Bsz*Csz*Hsz*Msz + 255)/256, 256, 0, stream>>>(x, regionA);
  dwt_cols <<<(2*M2*Csz*Bsz   + 255)/256, 256, 0, stream>>>(regionA, regionB);
  mix_wmma <<<4*M2, 128, 0, stream>>>(regionB, w1, w2, w3, w4, regionA);
  idwt_cols<<<(2*Bsz*Csz*Hsz*Msz + 255)/256, 256, 0, stream>>>(regionA, regionB);
  idwt_rows<<<(Bsz*Csz*Hsz*Wsz   + 255)/256, 256, 0, stream>>>(regionB, out);
}